// TGTEncoder_41068477285037
// MI455X (gfx1250) — compile-verified
//
#include <hip/hip_runtime.h>
#include <hip/hip_bf16.h>
#include <math.h>

// ---------------------------------------------------------------------------
// TGT encoder forward for MI455X (gfx1250, wave32, WMMA).
//
// GEMMs use V_WMMA_F32_16X16X4_F32 with LDS-staged weight panels:
//   block = 8 waves, 128x64 output tile; each wave = 16 rows x 4 col-tiles
//   (4 accumulators) so each A fragment feeds 4 WMMAs, and the W panel is
//   read from LDS (staged in 128-row / 32KB chunks), not re-fetched from
//   global per row-tile.  Out-of-range A rows are merely address-clamped:
//   WMMA row m of D depends only on row m of A, and those D rows are never
//   stored, so no zeroing (and no exec-masked loads) is needed.
//
// Input pointer layout assumption:
//   d_in[0..15]  : expr, token, path_feat, virt_feat, then 6 (src,dst) pairs
//                  in setup_inputs order.
//   d_in[16..]   : params pytree flattened in jax sorted-dict order:
//     for layer L in {0,1}:
//       for etype in sorted order [g2g,g2p,p2g,p2p,p2v,v2p]:
//         fields sorted: Wk,Wo,Wq,Wv,b,bk,bo,bq,bv,g        (10 each)
//     then: expr_b1, expr_b2, expr_w1, expr_w2, gene_b, gene_w,
//           path_b, path_w, token_emb, virt_b, virt_w        (11)
// ---------------------------------------------------------------------------

#define DIM   512
#define HEADS 8
#define DK    64
#define NG    20000
#define NP    2000
#define NVIRT 1
#define KCH   128     // K-chunk staged in LDS (128 rows x 64 cols = 32KB)

typedef __attribute__((ext_vector_type(2))) float v2f;
typedef __attribute__((ext_vector_type(8))) float v8f;

// ---- order-preserving float <-> uint encoding for atomicMax-based segment max
__device__ __forceinline__ unsigned fenc(float x) {
  unsigned u = __float_as_uint(x);
  return (u & 0x80000000u) ? ~u : (u | 0x80000000u);
}
__device__ __forceinline__ float fdec(unsigned e) {
  unsigned u = (e & 0x80000000u) ? (e & 0x7FFFFFFFu) : ~e;
  return __uint_as_float(u);
}

// ---------------------------------------------------------------------------
// C[M x N] = A[M x K] * W[K x N] + bias   (row-major, ldc-strided output)
// Grid: (N/64, ceil(M/128)); block 256 threads = 8 waves.
// Wave w: rows [blockIdx.y*128 + w*16, +16), cols [blockIdx.x*64, +64).
// WMMA fp32 16x16x4.  A frag: lanes 0-15 rows, VGPR0/1 = K,K+1; lanes 16-31
// same rows, K+2,K+3.  B mirrored with N across lanes.  C/D: lane<16 rows
// 0-7 (VGPR g = row g), lane>=16 rows 8-15; col = lane&15.
// ---------------------------------------------------------------------------
__global__ __launch_bounds__(256)
void gemm_bias_wmma(const float* __restrict__ A, int lda,
                    const float* __restrict__ W,
                    const float* __restrict__ bias,
                    float* __restrict__ C, int ldc,
                    int M, int K, int N) {
  __shared__ float ldsW[KCH * 64];

  const int tid  = threadIdx.x;
  const int wave = tid >> 5;
  const int lane = tid & 31;
  const int row0 = blockIdx.y * 128 + wave * 16;
  const int col0 = blockIdx.x * 64;

  const int l15   = lane & 15;
  const int khalf = (lane >> 4) << 1;        // 0 or 2
  const int arow  = row0 + l15;
  // clamp row so loads are always in-bounds; garbage rows feed only D rows
  // that the guarded epilogue never stores.
  const int arow_c = (arow < M) ? arow : (M - 1);
  const float* Arow = A + (size_t)arow_c * lda + khalf;

  v8f acc[4] = {{}, {}, {}, {}};

  for (int kc = 0; kc < K; kc += KCH) {
    // ---- stage W[kc..kc+KCH) x [col0..col0+64) into LDS (float4 per thread)
    for (int idx = tid; idx < KCH * 16; idx += 256) {
      const int kr = idx >> 4;               // row within chunk
      const int q  = idx & 15;               // float4 within 64-wide row
      const float* wp = W + (size_t)(kc + kr) * N + col0 + q * 4;
      *(float4*)(ldsW + kr * 64 + q * 4) = *(const float4*)wp;
      if (kc + KCH < K) __builtin_prefetch(wp + (size_t)KCH * N, 0, 1);
    }
    __syncthreads();

    // ---- 32 K-steps of 4: 1 unconditional global b64 (A, prefetched one
    //      step ahead) + 4 LDS B frags -> 4 WMMA
    const float* apc = Arow + kc;
    v2f a;
    a[0] = apc[0];
    a[1] = apc[1];
    for (int k0 = 0; k0 < KCH; k0 += 4) {
      const int kn = (k0 + 4 < KCH) ? (k0 + 4) : k0;  // clamped prefetch idx
      v2f an;
      an[0] = apc[kn];
      an[1] = apc[kn + 1];

      const float* bp = ldsW + (k0 + khalf) * 64 + l15;
#pragma unroll
      for (int ct = 0; ct < 4; ++ct) {
        v2f b;
        b[0] = bp[ct * 16];
        b[1] = bp[ct * 16 + 64];
        acc[ct] = __builtin_amdgcn_wmma_f32_16x16x4_f32(
            false, a, false, b, (short)0, acc[ct], false, false);
      }
      a = an;
    }
    __syncthreads();
  }

  // ---- epilogue: bias + store
  const int rbase = row0 + ((lane >> 4) ? 8 : 0);
#pragma unroll
  for (int ct = 0; ct < 4; ++ct) {
    const int col = col0 + ct * 16 + l15;
    const float bv = bias[col];
#pragma unroll
    for (int g = 0; g < 8; ++g) {
      const int r = rbase + g;
      if (r < M) C[(size_t)r * ldc + col] = acc[ct][g] + bv;
    }
  }
}

// ---------------------------------------------------------------------------
__global__ void fill_u32_kernel(unsigned* __restrict__ p, unsigned v, long n) {
  long t = (long)blockIdx.x * blockDim.x + threadIdx.x;
  if (t < n) p[t] = v;
}

// E1[i][j] = gelu(expr[i] * w1[j] + b1[j])   (exact gelu, erf form)
__global__ void expr_mlp1_kernel(const float* __restrict__ expr,
                                 const float* __restrict__ w1,
                                 const float* __restrict__ b1,
                                 float* __restrict__ E1) {
  int t = blockIdx.x * blockDim.x + threadIdx.x;
  if (t >= NG * 256) return;
  int i = t >> 8, j = t & 255;
  float x = expr[i] * w1[j] + b1[j];
  E1[t] = 0.5f * x * (1.0f + erff(x * 0.70710678118654752440f));
}

// X[i][256+j] = token_emb[token[i]][j]
__global__ void token_gather_kernel(const float* __restrict__ emb,
                                    const int* __restrict__ token,
                                    float* __restrict__ X) {
  int t = blockIdx.x * blockDim.x + threadIdx.x;
  if (t >= NG * 256) return;
  int i = t >> 8, j = t & 255;
  X[(size_t)i * DIM + 256 + j] = emb[(size_t)token[i] * 256 + j];
}

// out[i][j] = feat[i] * w[j] + b[j]
__global__ void bcast_embed_kernel(const float* __restrict__ feat,
                                   const float* __restrict__ w,
                                   const float* __restrict__ b,
                                   float* __restrict__ out, int n) {
  int t = blockIdx.x * blockDim.x + threadIdx.x;
  if (t >= n * DIM) return;
  int i = t >> 9, j = t & 511;
  out[t] = feat[i] * w[j] + b[j];
}

// score[e][h] = (K[src[e],h,:] . Q[dst[e],h,:]) / 8 ; atomicMax into smax
__global__ void edge_score_kernel(const float* __restrict__ Q,
                                  const float* __restrict__ K,
                                  const int* __restrict__ src,
                                  const int* __restrict__ dst,
                                  float* __restrict__ score,
                                  unsigned* __restrict__ smax,
                                  int n_edge) {
  int t = blockIdx.x * blockDim.x + threadIdx.x;
  if (t >= n_edge * HEADS) return;
  int e = t >> 3, h = t & 7;
  int s = src[e], d = dst[e];
  const float4* kp = (const float4*)(K + (size_t)s * DIM + h * DK);
  const float4* qp = (const float4*)(Q + (size_t)d * DIM + h * DK);
  float acc = 0.0f;
#pragma unroll
  for (int i = 0; i < DK / 4; ++i) {
    float4 kv = kp[i], qv = qp[i];
    acc += kv.x * qv.x + kv.y * qv.y + kv.z * qv.z + kv.w * qv.w;
  }
  acc *= 0.125f;  // 1/sqrt(64)
  score[t] = acc;
  atomicMax(&smax[d * HEADS + h], fenc(acc));
}

// score[e][h] = exp(score - smax[dst]);  z[dst][h] += score
__global__ void edge_exp_kernel(float* __restrict__ score,
                                const unsigned* __restrict__ smax,
                                float* __restrict__ z,
                                const int* __restrict__ dst, int n_edge) {
  int t = blockIdx.x * blockDim.x + threadIdx.x;
  if (t >= n_edge * HEADS) return;
  int e = t >> 3, h = t & 7;
  int d = dst[e];
  float ev = expf(score[t] - fdec(smax[d * HEADS + h]));
  score[t] = ev;
  atomicAdd(&z[d * HEADS + h], ev);
}

// Agg[dst][c..c+3] += V[src][c..c+3] * score[e][c>>6] / z[dst][c>>6]
// one thread per (edge, 4-dim chunk): float4 V read, 4 atomics
__global__ void edge_agg_kernel(const float* __restrict__ score,
                                const float* __restrict__ z,
                                const float* __restrict__ V,
                                const int* __restrict__ src,
                                const int* __restrict__ dst,
                                float* __restrict__ Agg, int n_edge) {
  long t = (long)blockIdx.x * blockDim.x + threadIdx.x;
  if (t >= (long)n_edge * (DIM / 4)) return;
  int e  = (int)(t >> 7);
  int qi = (int)(t & 127);          // float4 index within row
  int c  = qi * 4;
  int h  = qi >> 4;                 // 16 float4s per 64-wide head
  int sn = src[e], dn = dst[e];
  float a = score[e * HEADS + h] / z[dn * HEADS + h];
  float4 v4 = *(const float4*)(V + (size_t)sn * DIM + c);
  float* ap = Agg + (size_t)dn * DIM + c;
  atomicAdd(ap + 0, v4.x * a);
  atomicAdd(ap + 1, v4.y * a);
  atomicAdd(ap + 2, v4.z * a);
  atomicAdd(ap + 3, v4.w * a);
}

// h[i] = LayerNorm(h[i] + proj[i]) * g + b     (one wave32 per 512-wide row)
__global__ void residual_ln_kernel(float* __restrict__ h,
                                   const float* __restrict__ proj,
                                   const float* __restrict__ g,
                                   const float* __restrict__ b, int n) {
  int row  = blockIdx.x * (blockDim.x >> 5) + (threadIdx.x >> 5);
  int lane = threadIdx.x & 31;
  if (row >= n) return;
  float* hp = h + (size_t)row * DIM;
  const float* pp = proj + (size_t)row * DIM;
  float vals[DIM / 32];
  float s = 0.0f;
#pragma unroll
  for (int i = 0; i < DIM / 32; ++i) {
    int c = lane + i * 32;
    float x = hp[c] + pp[c];
    vals[i] = x;
    s += x;
  }
#pragma unroll
  for (int off = 16; off > 0; off >>= 1) s += __shfl_down(s, off);
  s = __shfl(s, 0);
  float mean = s * (1.0f / DIM);
  float v = 0.0f;
#pragma unroll
  for (int i = 0; i < DIM / 32; ++i) {
    float d0 = vals[i] - mean;
    v += d0 * d0;
  }
#pragma unroll
  for (int off = 16; off > 0; off >>= 1) v += __shfl_down(v, off);
  v = __shfl(v, 0);
  float inv = rsqrtf(v * (1.0f / DIM) + 1e-5f);
#pragma unroll
  for (int i = 0; i < DIM / 32; ++i) {
    int c = lane + i * 32;
    hp[c] = g[c] * (vals[i] - mean) * inv + b[c];
  }
}

// ---------------------------------------------------------------------------
struct LayerP {
  const float *Wq, *bq, *Wk, *bk, *Wv, *bv, *Wo, *bo, *g, *b;
};

extern "C" void kernel_launch(void* const* d_in, const int* in_sizes, int n_in,
                              void* d_out, int out_size, void* d_ws, size_t ws_size,
                              hipStream_t stream) {
  (void)in_sizes; (void)n_in; (void)out_size; (void)ws_size;

  // ---- inputs -------------------------------------------------------------
  const float* expr      = (const float*)d_in[0];
  const int*   token     = (const int*)  d_in[1];
  const float* path_feat = (const float*)d_in[2];
  const float* virt_feat = (const float*)d_in[3];
  const int* esrc[6] = { (const int*)d_in[4],  (const int*)d_in[6],
                         (const int*)d_in[8],  (const int*)d_in[10],
                         (const int*)d_in[12], (const int*)d_in[14] };
  const int* edst[6] = { (const int*)d_in[5],  (const int*)d_in[7],
                         (const int*)d_in[9],  (const int*)d_in[11],
                         (const int*)d_in[13], (const int*)d_in[15] };
  // edge type order here matches setup_inputs: g2g,g2p,p2g,p2p,v2p,p2v
  const int ecount[6] = { 400000, 200000, 200000, 40000, 2000, 2000 };

  // params: blocks first (sorted etype order g2g,g2p,p2g,p2p,p2v,v2p), then tail
  const int PB = 16;
  auto layerP = [&](int L, int et_sorted) -> LayerP {
    const int base = PB + L * 60 + et_sorted * 10;
    LayerP P;
    P.Wk = (const float*)d_in[base + 0];
    P.Wo = (const float*)d_in[base + 1];
    P.Wq = (const float*)d_in[base + 2];
    P.Wv = (const float*)d_in[base + 3];
    P.b  = (const float*)d_in[base + 4];
    P.bk = (const float*)d_in[base + 5];
    P.bo = (const float*)d_in[base + 6];
    P.bq = (const float*)d_in[base + 7];
    P.bv = (const float*)d_in[base + 8];
    P.g  = (const float*)d_in[base + 9];
    return P;
  };
  const int TB = PB + 120;
  const float* expr_b1   = (const float*)d_in[TB + 0];
  const float* expr_b2   = (const float*)d_in[TB + 1];
  const float* expr_w1   = (const float*)d_in[TB + 2];
  const float* expr_w2   = (const float*)d_in[TB + 3];
  const float* gene_b    = (const float*)d_in[TB + 4];
  const float* gene_w    = (const float*)d_in[TB + 5];
  const float* path_b    = (const float*)d_in[TB + 6];
  const float* path_w    = (const float*)d_in[TB + 7];
  const float* token_emb = (const float*)d_in[TB + 8];
  const float* virt_b    = (const float*)d_in[TB + 9];
  const float* virt_w    = (const float*)d_in[TB + 10];

  // ---- workspace layout (floats, 256B aligned) ----------------------------
  size_t off = 0;
  auto alloc = [&](size_t nf) -> float* {
    float* p = (float*)d_ws + off;
    off += (nf + 63) & ~(size_t)63;
    return p;
  };
  float* hg   = alloc((size_t)NG * DIM);
  float* hp   = alloc((size_t)NP * DIM);
  float* hv   = alloc(DIM);
  float* Qb   = alloc((size_t)NG * DIM);
  float* Kb   = alloc((size_t)NG * DIM);
  float* Vb   = alloc((size_t)NG * DIM);
  float* Agg  = alloc((size_t)NG * DIM);
  float* Tmp  = alloc((size_t)NG * DIM);   // concat buffer + proj output
  float* score = alloc((size_t)400000 * HEADS);
  unsigned* smax = (unsigned*)alloc((size_t)NG * HEADS);
  float* zsum = alloc((size_t)NG * HEADS);
  float* E1   = Agg;                        // reuse: Agg idle during embedding

  auto gemm = [&](const float* A, int lda, const float* W, const float* bias,
                  float* C, int ldc, int M, int K, int N) {
    dim3 grid((N + 63) / 64, (M + 127) / 128);
    gemm_bias_wmma<<<grid, 256, 0, stream>>>(A, lda, W, bias, C, ldc, M, K, N);
  };

  // ---- embeddings ---------------------------------------------------------
  {
    int n = NG * 256;
    expr_mlp1_kernel<<<(n + 255) / 256, 256, 0, stream>>>(expr, expr_w1, expr_b1, E1);
    gemm(E1, 256, expr_w2, expr_b2, Tmp, DIM, NG, 256, 256);   // cols 0..255 of X
    token_gather_kernel<<<(n + 255) / 256, 256, 0, stream>>>(token_emb, token, Tmp);
    gemm(Tmp, DIM, gene_w, gene_b, hg, DIM, NG, DIM, DIM);
    int np = NP * DIM;
    bcast_embed_kernel<<<(np + 255) / 256, 256, 0, stream>>>(path_feat, path_w, path_b, hp, NP);
    int nv = NVIRT * DIM;
    bcast_embed_kernel<<<(nv + 255) / 256, 256, 0, stream>>>(virt_feat, virt_w, virt_b, hv, NVIRT);
  }

  // ---- graph transformer layer --------------------------------------------
  auto gt = [&](const LayerP& p, const float* hsrc, int n_src, float* hdst,
                int n_dst, const int* src, const int* dst, int n_edge) {
    gemm(hdst, DIM, p.Wq, p.bq, Qb, DIM, n_dst, DIM, DIM);
    gemm(hsrc, DIM, p.Wk, p.bk, Kb, DIM, n_src, DIM, DIM);
    gemm(hsrc, DIM, p.Wv, p.bv, Vb, DIM, n_src, DIM, DIM);

    long nsm = (long)n_dst * HEADS;
    fill_u32_kernel<<<(nsm + 255) / 256, 256, 0, stream>>>(smax, 0u, nsm);
    long neh = (long)n_edge * HEADS;
    edge_score_kernel<<<(neh + 255) / 256, 256, 0, stream>>>(Qb, Kb, src, dst, score, smax, n_edge);
    fill_u32_kernel<<<(nsm + 255) / 256, 256, 0, stream>>>((unsigned*)zsum, 0u, nsm);
    edge_exp_kernel<<<(neh + 255) / 256, 256, 0, stream>>>(score, smax, zsum, dst, n_edge);
    long nagg = (long)n_dst * DIM;
    fill_u32_kernel<<<(nagg + 255) / 256, 256, 0, stream>>>((unsigned*)Agg, 0u, nagg);
    long neq = (long)n_edge * (DIM / 4);
    edge_agg_kernel<<<(int)((neq + 255) / 256), 256, 0, stream>>>(score, zsum, Vb, src, dst, Agg, n_edge);

    gemm(Agg, DIM, p.Wo, p.bo, Tmp, DIM, n_dst, DIM, DIM);
    long nw = (long)n_dst * 32;
    residual_ln_kernel<<<(nw + 255) / 256, 256, 0, stream>>>(hdst, Tmp, p.g, p.b, n_dst);
  };

  // sorted-param etype index for each processing step:
  //   processing: g2g,g2p,p2g,p2p,v2p,p2v  ->  sorted: 0,1,2,3,5,4
  for (int L = 0; L < 2; ++L) {
    gt(layerP(L, 0), hg, NG,    hg, NG,    esrc[0], edst[0], ecount[0]);  // g2g
    gt(layerP(L, 1), hg, NG,    hp, NP,    esrc[1], edst[1], ecount[1]);  // g2p
    gt(layerP(L, 2), hp, NP,    hg, NG,    esrc[2], edst[2], ecount[2]);  // p2g
    gt(layerP(L, 3), hp, NP,    hp, NP,    esrc[3], edst[3], ecount[3]);  // p2p
    gt(layerP(L, 5), hv, NVIRT, hp, NP,    esrc[4], edst[4], ecount[4]);  // v2p
    gt(layerP(L, 4), hp, NP,    hv, NVIRT, esrc[5], edst[5], ecount[5]);  // p2v
  }

  // ---- outputs: concat(hg, hp, hv) ----------------------------------------
  float* out = (float*)d_out;
  hipMemcpyAsync(out,                       hg, (size_t)NG * DIM * 4, hipMemcpyDeviceToDevice, stream);
  hipMemcpyAsync(out + (size_t)NG * DIM,    hp, (size_t)NP * DIM * 4, hipMemcpyDeviceToDevice, stream);
  hipMemcpyAsync(out + (size_t)(NG + NP) * DIM, hv, (size_t)DIM * 4,  hipMemcpyDeviceToDevice, stream);
}